// JointEmbeddingClassifier_66769561584332
// MI455X (gfx1250) — compile-verified
//
#include <hip/hip_runtime.h>

typedef float v2f __attribute__((ext_vector_type(2)));
typedef float v8f __attribute__((ext_vector_type(8)));

#define B_SZ   4096
#define D_INN  4096
#define D_PROJ 1024
#define N_CLSS 32
#define NC0    32
#define NC1    64

// ---------------- Kernel 1: fused dual projection GEMM ----------------
// Virtual GEMM: [4096 x 4096] @ [4096 x 2048]  where cols [0,1024) = Wp -> parent_proj,
// cols [1024,2048) = Ws -> child_proj.  BM=64 BN=64 BK=16, 256 threads = 8 waves,
// each wave owns one 16-row strip and two 16x16 f32 WMMA accumulators.
// LDS is double-buffered; B tile is stored K-transposed so both A and B WMMA
// fragments are single aligned ds_load_b64's.
#define BM 64
#define BN 64
#define BK 16
#define LDT 18   // padded tile row (floats): 72B rows -> b64-aligned, conflict-light

__global__ __launch_bounds__(256)
void proj_gemm_kernel(const float* __restrict__ x,
                      const float* __restrict__ Wp, const float* __restrict__ bp,
                      const float* __restrict__ Ws, const float* __restrict__ bs,
                      float* __restrict__ Pout, float* __restrict__ Sout)
{
    __shared__ float As[2][BM * LDT];   // As[buf][m*LDT + k]
    __shared__ float Bt[2][BN * LDT];   // Bt[buf][n*LDT + k]  (transposed W tile)

    const int bm  = blockIdx.x * BM;
    const int gn  = blockIdx.y * BN;        // fused-N base in [0, 2048)
    const int tid = threadIdx.x;
    const int wave = tid >> 5;
    const int lane = tid & 31;

    const bool is_parent = (gn < D_PROJ);
    const float* __restrict__ W    = is_parent ? Wp : Ws;
    const float* __restrict__ bias = is_parent ? bp : bs;
    float* __restrict__ Cout       = is_parent ? Pout : Sout;
    const int cn = is_parent ? gn : (gn - D_PROJ);   // column base within this output

    const int m16 = (wave & 3) * 16;        // wave's M subtile
    const int n16 = (wave >> 2) * 32;       // wave's N subtile pair base

    v8f acc0 = {0.f,0.f,0.f,0.f,0.f,0.f,0.f,0.f};
    v8f acc1 = {0.f,0.f,0.f,0.f,0.f,0.f,0.f,0.f};

    // per-lane WMMA fragment coords (16x16x4 f32 layout, ISA 7.12.2)
    const int fm = lane & 15;               // M (for A) / N (for B,C)
    const int fk = (lane >> 4) * 2;         // K pair base {0,2}

    // cooperative staging assignments (one float4 per thread per tile)
    const int arow = tid >> 2;              // 0..63  (M row of A tile)
    const int acol = (tid & 3) * 4;         // 0,4,8,12 (K col of A tile)
    const int brow = tid >> 4;              // 0..15  (K row of W tile)
    const int bcol = (tid & 15) * 4;        // 0..60  (N col of W tile)

    const float* xg = x + (size_t)(bm + arow) * D_INN + acol;
    const float* wg = W + (size_t)brow * D_PROJ + cn + bcol;

    // ---- prologue: stage tile 0 into buffer 0 ----
    {
        float4 av = *(const float4*)(xg);
        float4 bv = *(const float4*)(wg);
        As[0][arow * LDT + acol + 0] = av.x;
        As[0][arow * LDT + acol + 1] = av.y;
        As[0][arow * LDT + acol + 2] = av.z;
        As[0][arow * LDT + acol + 3] = av.w;
        Bt[0][(bcol + 0) * LDT + brow] = bv.x;
        Bt[0][(bcol + 1) * LDT + brow] = bv.y;
        Bt[0][(bcol + 2) * LDT + brow] = bv.z;
        Bt[0][(bcol + 3) * LDT + brow] = bv.w;
    }
    __syncthreads();

    int buf = 0;
    for (int kb = 0; kb < D_INN; kb += BK) {
        const int nxt = buf ^ 1;
        // stage tile kb+BK into the other buffer while computing on `buf`
        if (kb + BK < D_INN) {
            float4 av = *(const float4*)(xg + kb + BK);
            float4 bv = *(const float4*)(wg + (size_t)(kb + BK) * D_PROJ);
            if (kb + 2 * BK < D_INN) {
                __builtin_prefetch(xg + kb + 2 * BK, 0, 1);
                __builtin_prefetch(wg + (size_t)(kb + 2 * BK) * D_PROJ, 0, 1);
            }
            As[nxt][arow * LDT + acol + 0] = av.x;
            As[nxt][arow * LDT + acol + 1] = av.y;
            As[nxt][arow * LDT + acol + 2] = av.z;
            As[nxt][arow * LDT + acol + 3] = av.w;
            Bt[nxt][(bcol + 0) * LDT + brow] = bv.x;
            Bt[nxt][(bcol + 1) * LDT + brow] = bv.y;
            Bt[nxt][(bcol + 2) * LDT + brow] = bv.z;
            Bt[nxt][(bcol + 3) * LDT + brow] = bv.w;
        }

        #pragma unroll
        for (int kk = 0; kk < BK; kk += 4) {
            v2f a  = *(const v2f*)&As[buf][(m16 + fm) * LDT + kk + fk];       // ds_load_b64
            v2f b0 = *(const v2f*)&Bt[buf][(n16 + fm) * LDT + kk + fk];       // ds_load_b64
            v2f b1 = *(const v2f*)&Bt[buf][(n16 + 16 + fm) * LDT + kk + fk];  // ds_load_b64
            acc0 = __builtin_amdgcn_wmma_f32_16x16x4_f32(false, a, false, b0,
                                                         (short)0, acc0, false, false);
            acc1 = __builtin_amdgcn_wmma_f32_16x16x4_f32(false, a, false, b1,
                                                         (short)0, acc1, false, false);
        }
        __syncthreads();   // publishes buf^1, protects buf from next overwrite
        buf = nxt;
    }

    // epilogue: acc[i] = C[m16 + i + 8*(lane/16)][n16 + fm]
    const int col   = cn + n16 + fm;
    const int rbase = bm + m16 + (lane >> 4) * 8;
    const float bv0 = bias[col];
    const float bv1 = bias[col + 16];
    #pragma unroll
    for (int i = 0; i < 8; ++i) {
        const size_t r = (size_t)(rbase + i);
        Cout[r * D_PROJ + col]      = acc0[i] + bv0;
        Cout[r * D_PROJ + col + 16] = acc1[i] + bv1;
    }
}

// ---------------- Kernel 2: parent logits = P @ Wc + bc ----------------
// [4096 x 1024] @ [1024 x 32]. Wc is 128 KB -> L2 resident; P streams from L2/HBM.
// 8 waves per block, each wave owns a 16-row strip and both 16-col tiles.
__global__ __launch_bounds__(256)
void parent_logits_kernel(const float* __restrict__ P,
                          const float* __restrict__ Wc,
                          const float* __restrict__ bc,
                          float* __restrict__ out)
{
    const int wave = threadIdx.x >> 5;
    const int lane = threadIdx.x & 31;
    const int m0 = (blockIdx.x * 8 + wave) * 16;
    const int fm = lane & 15;
    const int fk = (lane >> 4) * 2;

    v8f acc0 = {0.f,0.f,0.f,0.f,0.f,0.f,0.f,0.f};
    v8f acc1 = {0.f,0.f,0.f,0.f,0.f,0.f,0.f,0.f};

    const float* prow = P + (size_t)(m0 + fm) * D_PROJ;
    for (int k = 0; k < D_PROJ; k += 4) {
        v2f a = *(const v2f*)(prow + k + fk);
        v2f b0, b1;
        b0.x = Wc[(size_t)(k + fk)     * N_CLSS + fm];
        b0.y = Wc[(size_t)(k + fk + 1) * N_CLSS + fm];
        b1.x = Wc[(size_t)(k + fk)     * N_CLSS + 16 + fm];
        b1.y = Wc[(size_t)(k + fk + 1) * N_CLSS + 16 + fm];
        acc0 = __builtin_amdgcn_wmma_f32_16x16x4_f32(false, a, false, b0,
                                                     (short)0, acc0, false, false);
        acc1 = __builtin_amdgcn_wmma_f32_16x16x4_f32(false, a, false, b1,
                                                     (short)0, acc1, false, false);
    }

    const int rbase = m0 + (lane >> 4) * 8;
    const float b0v = bc[fm];
    const float b1v = bc[16 + fm];
    #pragma unroll
    for (int i = 0; i < 8; ++i) {
        const size_t r = (size_t)(rbase + i);
        out[r * N_CLSS + fm]      = acc0[i] + b0v;
        out[r * N_CLSS + 16 + fm] = acc1[i] + b1v;
    }
}

// ---------------- Kernel 3: routed child logits ----------------
// One block per sample. Stage child_proj[b] (4KB) in LDS; lane-per-output matvec
// against the label-selected expert bank (12 MB total -> L2 resident).
__global__ __launch_bounds__(128)
void child_logits_kernel(const float* __restrict__ S, const int* __restrict__ y,
                         const float* __restrict__ cw0, const float* __restrict__ cb0,
                         const float* __restrict__ cw1, const float* __restrict__ cb1,
                         float* __restrict__ out0, float* __restrict__ out1)
{
    __shared__ float s[D_PROJ];
    const int b   = blockIdx.x;
    const int tid = threadIdx.x;

    const float4* src = (const float4*)(S + (size_t)b * D_PROJ);
    float4* dst = (float4*)s;
    #pragma unroll
    for (int i = 0; i < D_PROJ / 4 / 128; ++i)
        dst[tid + i * 128] = src[tid + i * 128];
    __syncthreads();

    const int c = y[b];
    if (tid < 32) {                               // wave 0: level-0 experts (32 outs)
        const float* w = cw0 + ((size_t)c * NC0 + tid) * D_PROJ;
        float acc = 0.f;
        for (int d = 0; d < D_PROJ; d += 4) {
            float4 wv = *(const float4*)(w + d);
            acc += s[d] * wv.x + s[d+1] * wv.y + s[d+2] * wv.z + s[d+3] * wv.w;
        }
        out0[(size_t)b * NC0 + tid] = acc + cb0[(size_t)c * NC0 + tid];
    } else if (tid < 96) {                        // waves 1-2: level-1 experts (64 outs)
        const int n = tid - 32;
        const float* w = cw1 + ((size_t)c * NC1 + n) * D_PROJ;
        float acc = 0.f;
        for (int d = 0; d < D_PROJ; d += 4) {
            float4 wv = *(const float4*)(w + d);
            acc += s[d] * wv.x + s[d+1] * wv.y + s[d+2] * wv.z + s[d+3] * wv.w;
        }
        out1[(size_t)b * NC1 + n] = acc + cb1[(size_t)c * NC1 + n];
    }
}

extern "C" void kernel_launch(void* const* d_in, const int* in_sizes, int n_in,
                              void* d_out, int out_size, void* d_ws, size_t ws_size,
                              hipStream_t stream) {
    const float* x   = (const float*)d_in[0];
    const int*   y   = (const int*)  d_in[1];
    const float* Wp  = (const float*)d_in[2];
    const float* bp  = (const float*)d_in[3];
    const float* Ws  = (const float*)d_in[4];
    const float* bs  = (const float*)d_in[5];
    const float* Wc  = (const float*)d_in[6];
    const float* bc  = (const float*)d_in[7];
    const float* cw0 = (const float*)d_in[8];
    const float* cb0 = (const float*)d_in[9];
    const float* cw1 = (const float*)d_in[10];
    const float* cb1 = (const float*)d_in[11];

    float* out           = (float*)d_out;
    float* parent_logits = out;                                   // [4096, 32]
    float* child0        = parent_logits + (size_t)B_SZ * N_CLSS; // [4096, 32]
    float* child1        = child0 + (size_t)B_SZ * NC0;           // [4096, 64]
    float* Pproj         = child1 + (size_t)B_SZ * NC1;           // [4096, 1024]
    float* Sproj         = Pproj + (size_t)B_SZ * D_PROJ;         // [4096, 1024]

    dim3 g1(B_SZ / BM, (2 * D_PROJ) / BN);   // 64 x 32 blocks
    proj_gemm_kernel<<<g1, 256, 0, stream>>>(x, Wp, bp, Ws, bs, Pproj, Sproj);
    parent_logits_kernel<<<B_SZ / 128, 256, 0, stream>>>(Pproj, Wc, bc, parent_logits);
    child_logits_kernel<<<B_SZ, 128, 0, stream>>>(Sproj, y, cw0, cb0, cw1, cb1,
                                                  child0, child1);
}